// LSTMClassifier_66365834658103
// MI455X (gfx1250) — compile-verified
//
#include <hip/hip_runtime.h>
#include <math.h>

// ---------- CDNA5 WMMA f32 16x16x4 ----------
typedef __attribute__((ext_vector_type(2))) float v2f;
typedef __attribute__((ext_vector_type(8))) float v8f;

__device__ __forceinline__ v8f wmma4(v2f a, v2f b, v8f c) {
  // D = A(16x4) * B(4x16) + C(16x16), full fp32
  return __builtin_amdgcn_wmma_f32_16x16x4_f32(
      false, a, false, b, (short)0, c, false, false);
}

__device__ __forceinline__ float sigmoidf_(float x) {
  return 1.0f / (1.0f + __expf(-x));
}

// ---------- constants ----------
#define BATCH 64
#define SEQ 3072
#define DIN 128
#define EMB 64
#define HID 256
#define NOUT 10
#define TOUT 1024
#define LSTM_BLOCKS 16

// ---------- 1) per-(b,s) inverse L2 norm (float4 coalesced) ----------
__global__ void __launch_bounds__(256) norm_kernel(
    const float* __restrict__ x, float* __restrict__ inv_norm) {
  int row = blockIdx.x * (blockDim.x >> 5) + (threadIdx.x >> 5);
  int lane = threadIdx.x & 31;
  const float4* p = (const float4*)(x + (size_t)row * DIN);
  float4 v = p[lane];
  float ss = v.x * v.x + v.y * v.y + v.z * v.z + v.w * v.w;
  #pragma unroll
  for (int off = 16; off; off >>= 1) ss += __shfl_xor(ss, off, 32);
  if (lane == 0) inv_norm[row] = 1.0f / fmaxf(sqrtf(ss), 1e-12f);
}

// ---------- 2) fused normalize + conv1d(stride3,k3) + bias + relu as WMMA GEMM
//  A = [B*T, 384] (row (b,t) = contiguous slice of input)
//  Bm pair layout: [(k/2)][e][2]  ->  per-lane fragment = one 8B LDS load
//  output act layout: [T][B][E]
__global__ void __launch_bounds__(128) conv_kernel(
    const float* __restrict__ x, const float* __restrict__ inv_norm,
    const float* __restrict__ conv_w, const float* __restrict__ conv_b,
    float* __restrict__ act) {
  extern __shared__ float Bm[];  // 384*64 floats, pair-interleaved
  for (int idx = threadIdx.x; idx < 384 * 64; idx += blockDim.x) {
    int r = idx & 1, e = (idx >> 1) & 63, p = idx >> 7;
    int j = 2 * p + r;  // k index in [0,384)
    Bm[idx] = conv_w[e * 384 + (j & 127) * 3 + (j >> 7)];
  }
  __syncthreads();

  const int wave = threadIdx.x >> 5, lane = threadIdx.x & 31;
  const int nn = lane & 15, hi = lane >> 4, kq = hi << 1;
  const int mtile = blockIdx.x * 4 + wave;   // 4096 m-tiles total
  const int mrow = mtile * 16 + nn;          // this lane's A row
  const int b = mrow >> 10, t = mrow & 1023;
  const float* arow = x + ((size_t)b * SEQ + t * 3) * DIN;  // 384 contiguous
  const float* nrow = inv_norm + b * SEQ + t * 3;
  const v2f* bm2 = (const v2f*)Bm;

  v8f acc[4] = { {0,0,0,0,0,0,0,0}, {0,0,0,0,0,0,0,0},
                 {0,0,0,0,0,0,0,0}, {0,0,0,0,0,0,0,0} };
  #pragma unroll 4
  for (int k4 = 0; k4 < 384; k4 += 4) {
    int j0 = k4 + kq;                       // even -> 8B aligned
    v2f a = *(const v2f*)(arow + j0);
    a *= nrow[j0 >> 7];                     // fused L2-normalize
    const v2f* brow = bm2 + (j0 >> 1) * 64;
    #pragma unroll
    for (int nt = 0; nt < 4; ++nt) {
      v2f bb = brow[nt * 16 + nn];          // single ds_load_b64
      acc[nt] = wmma4(a, bb, acc[nt]);
    }
  }
  #pragma unroll
  for (int nt = 0; nt < 4; ++nt) {
    int n = nt * 16 + nn;
    float bias = conv_b[n];
    #pragma unroll
    for (int v = 0; v < 8; ++v) {
      int mm = mtile * 16 + v + hi * 8;     // output row (b,t)
      int ob = mm >> 10, ot = mm & 1023;
      float r = acc[nt][v] + bias;
      act[(size_t)ot * (BATCH * EMB) + ob * EMB + n] = r > 0.f ? r : 0.f;
    }
  }
}

// ---------- 3) persistent LSTM: 16 blocks x 128 threads, grid barrier / step
//  cell state lives entirely in registers (block-private columns)
__global__ void __launch_bounds__(128) lstm_kernel(
    const float* __restrict__ act,   // [T][64][64]
    const float* __restrict__ w_ih,  // [1024][64]
    const float* __restrict__ w_hh,  // [1024][256]
    const float* __restrict__ b_ih, const float* __restrict__ b_hh,
    float* __restrict__ h0, float* __restrict__ h1,
    unsigned* __restrict__ bar) {
  extern __shared__ float smem[];
  float* Bx   = smem;                  // 4*64*16 floats, pair-interleaved
  float* Bh   = Bx + 4 * 64 * 16;      // 4*256*16 floats, pair-interleaved
  float* bias = Bh + 4 * 256 * 16;     // [4][16]
  const int j0 = blockIdx.x * 16;      // owned h-columns [j0, j0+16)

  // stage weights in pair layout: addr = ((g*K/2 + k/2)*16 + n)*2 + (k&1)
  for (int idx = threadIdx.x; idx < 4 * 64 * 16; idx += blockDim.x) {
    int r = idx & 1, n = (idx >> 1) & 15, gp = idx >> 5;
    int p = gp & 31, g = gp >> 5, k = 2 * p + r;
    Bx[idx] = w_ih[(g * HID + j0 + n) * EMB + k];
  }
  for (int idx = threadIdx.x; idx < 4 * 256 * 16; idx += blockDim.x) {
    int r = idx & 1, n = (idx >> 1) & 15, gp = idx >> 5;
    int p = gp & 127, g = gp >> 7, k = 2 * p + r;
    Bh[idx] = w_hh[(g * HID + j0 + n) * HID + k];
  }
  if (threadIdx.x < 64) {
    int g = threadIdx.x >> 4, n = threadIdx.x & 15;
    bias[threadIdx.x] = b_ih[g * HID + j0 + n] + b_hh[g * HID + j0 + n];
  }
  __syncthreads();

  const int wave = threadIdx.x >> 5, lane = threadIdx.x & 31;
  const int nn = lane & 15, hi = lane >> 4, kq = hi << 1;
  const int m0 = wave * 16 + nn;       // A-fragment row (batch index)
  const int col = j0 + nn;             // this lane's h column
  const v2f* bx2 = (const v2f*)Bx;
  const v2f* bh2 = (const v2f*)Bh;

  const float bi = bias[0 * 16 + nn], bf = bias[1 * 16 + nn];
  const float bg = bias[2 * 16 + nn], bo = bias[3 * 16 + nn];

  float creg[8] = {0.f, 0.f, 0.f, 0.f, 0.f, 0.f, 0.f, 0.f};

  for (int s = 0; s < TOUT; ++s) {
    const float* hprev = (s & 1) ? h1 : h0;
    float* hnext = (s & 1) ? h0 : h1;
    const float* at = act + (size_t)s * (BATCH * EMB);

    // prefetch next timestep's x slice while this step works / spins
    if (s + 1 < TOUT)
      __builtin_prefetch(at + BATCH * EMB + m0 * EMB, 0, 1);

    v8f acc[4] = { {0,0,0,0,0,0,0,0}, {0,0,0,0,0,0,0,0},
                   {0,0,0,0,0,0,0,0}, {0,0,0,0,0,0,0,0} };
    // x_t @ w_ih^T contribution (K=64)
    #pragma unroll 4
    for (int k4 = 0; k4 < EMB; k4 += 4) {
      v2f a = *(const v2f*)(at + m0 * EMB + k4 + kq);
      int p = (k4 + kq) >> 1;
      #pragma unroll
      for (int g = 0; g < 4; ++g) {
        v2f bb = bx2[(g * 32 + p) * 16 + nn];
        acc[g] = wmma4(a, bb, acc[g]);
      }
    }
    // h @ w_hh^T contribution (K=256)
    #pragma unroll 4
    for (int k4 = 0; k4 < HID; k4 += 4) {
      v2f a = *(const v2f*)(hprev + m0 * HID + k4 + kq);
      int p = (k4 + kq) >> 1;
      #pragma unroll
      for (int g = 0; g < 4; ++g) {
        v2f bb = bh2[(g * 128 + p) * 16 + nn];
        acc[g] = wmma4(a, bb, acc[g]);
      }
    }
    // gate math on aligned fragments; c stays in registers
    #pragma unroll
    for (int v = 0; v < 8; ++v) {
      int m = wave * 16 + v + hi * 8;      // batch row
      float gi = sigmoidf_(acc[0][v] + bi);
      float gf = sigmoidf_(acc[1][v] + bf);
      float gg = tanhf(acc[2][v] + bg);
      float go = sigmoidf_(acc[3][v] + bo);
      float cn = gf * creg[v] + gi * gg;
      creg[v] = cn;
      hnext[m * HID + col] = go * tanhf(cn);
    }
    // device-wide barrier (monotonic counter in L2)
    __threadfence();
    __syncthreads();
    if (threadIdx.x == 0) {
      __hip_atomic_fetch_add(bar, 1u, __ATOMIC_ACQ_REL, __HIP_MEMORY_SCOPE_AGENT);
      unsigned target = (unsigned)(s + 1) * LSTM_BLOCKS;
      while (__hip_atomic_load(bar, __ATOMIC_ACQUIRE, __HIP_MEMORY_SCOPE_AGENT) < target) { }
    }
    __syncthreads();
    __threadfence();
  }
}

// ---------- 4) final linear [64,256] x [256,10] ----------
__global__ void __launch_bounds__(64) linear_kernel(
    const float* __restrict__ h, const float* __restrict__ lin_w,
    const float* __restrict__ lin_b, float* __restrict__ out) {
  int idx = blockIdx.x * blockDim.x + threadIdx.x;
  if (idx >= BATCH * NOUT) return;
  int b = idx / NOUT, o = idx % NOUT;
  float s = lin_b[o];
  for (int k = 0; k < HID; ++k) s += h[b * HID + k] * lin_w[o * HID + k];
  out[idx] = s;
}

extern "C" void kernel_launch(void* const* d_in, const int* in_sizes, int n_in,
                              void* d_out, int out_size, void* d_ws, size_t ws_size,
                              hipStream_t stream) {
  const float* x      = (const float*)d_in[0];
  // d_in[1] = r (unused), d_in[2] = batch_size (unused)
  const float* conv_w = (const float*)d_in[3];
  const float* conv_b = (const float*)d_in[4];
  const float* w_ih   = (const float*)d_in[5];
  const float* w_hh   = (const float*)d_in[6];
  const float* b_ih   = (const float*)d_in[7];
  const float* b_hh   = (const float*)d_in[8];
  const float* lin_w  = (const float*)d_in[9];
  const float* lin_b  = (const float*)d_in[10];
  float* out = (float*)d_out;

  char* ws = (char*)d_ws;
  float* inv_norm = (float*)ws;                              // 196608 f
  float* act      = (float*)(ws + 786432);                   // 4 Mi f (16 MB)
  float* h0       = (float*)(ws + 786432 + 16777216);        // 64x256
  float* h1       = h0 + BATCH * HID;
  unsigned* bar   = (unsigned*)(h1 + BATCH * HID);

  // zero h0,h1 and the barrier counter (ws is poisoned, not zeroed)
  hipMemsetAsync(h0, 0, (size_t)2 * BATCH * HID * sizeof(float) + 128, stream);

  norm_kernel<<<(BATCH * SEQ) / 8, 256, 0, stream>>>(x, inv_norm);
  conv_kernel<<<1024, 128, 384 * 64 * sizeof(float), stream>>>(
      x, inv_norm, conv_w, conv_b, act);
  lstm_kernel<<<LSTM_BLOCKS, 128,
                (4 * 64 * 16 + 4 * 256 * 16 + 64) * sizeof(float), stream>>>(
      act, w_ih, w_hh, b_ih, b_hh, h0, h1, bar);
  // final h lands in h0 (write buffer of step 1023 is (1023+1)&1 == 0)
  linear_kernel<<<(BATCH * NOUT + 63) / 64, 64, 0, stream>>>(h0, lin_w, lin_b, out);
}